// SelfAttentiveSpanExtractor_62938450755986
// MI455X (gfx1250) — compile-verified
//
#include <hip/hip_runtime.h>
#include <math.h>

// Problem constants (match reference)
#define B_  8
#define S_  2048
#define N_  512
#define D_  1024
#define MAXW 32
#define MASK_FILL (-1000.0f)

typedef float v2f __attribute__((ext_vector_type(2)));
typedef float v8f __attribute__((ext_vector_type(8)));

// ---------------------------------------------------------------------------
// Kernel 1: logits[b*S+s] = dot(seq[b,s,:], att_w) + att_b
// One wave32 computes 16 rows via V_WMMA_F32_16X16X4_F32, accumulating over
// K=1024 in chunks of 4. Full fp32 (matches reference dtype).
//
// A (16x4 f32) layout: lanes 0-15 -> M=lane, K=0(v0),1(v1);
//                      lanes 16-31 -> M=lane-16, K=2(v0),3(v1)
//   => each lane loads float2 at seq[row, k0 + (lane>=16 ? 2 : 0)]
// B (4x16 f32) layout: v0 = B[K=0,N=lane] | B[K=2,N=lane-16]
//                      v1 = B[K=1,N=lane] | B[K=3,N=lane-16]
//   => only column N=0 carries att_w: lane 0 gets (w[k0],w[k0+1]),
//      lane 16 gets (w[k0+2],w[k0+3]); all other lanes 0.
// D (16x16 f32): column N=0 => lane 0 holds M=0..7 in c[0..7],
//                              lane 16 holds M=8..15 in c[0..7].
// ---------------------------------------------------------------------------
__global__ __launch_bounds__(256)
void span_logits_wmma_kernel(const float* __restrict__ seq,
                             const float* __restrict__ att_w,
                             const float* __restrict__ att_b,
                             float* __restrict__ logits) {
    const int lane  = threadIdx.x & 31;
    const int wid   = threadIdx.x >> 5;
    const int rows0 = (blockIdx.x * 8 + wid) * 16;     // 16 rows per wave
    const int m     = lane & 15;
    const int koff  = (lane & 16) ? 2 : 0;

    const float* rowp = seq + (size_t)(rows0 + m) * D_ + koff;

    v8f c = {};
    #pragma unroll 8
    for (int k0 = 0; k0 < D_; k0 += 4) {
        float2 av = *(const float2*)(rowp + k0);           // 8B aligned
        float2 wv = *(const float2*)(att_w + k0 + koff);   // 8B aligned
        v2f a, b;
        a.x = av.x;
        a.y = av.y;
        b.x = (m == 0) ? wv.x : 0.0f;   // column N=0 only
        b.y = (m == 0) ? wv.y : 0.0f;
        c = __builtin_amdgcn_wmma_f32_16x16x4_f32(
                /*neg_a=*/false, a, /*neg_b=*/false, b,
                /*c_mod=*/(short)0, c, /*reuse_a=*/false, /*reuse_b=*/false);
    }

    const float bias = att_b[0];
    if (m == 0) {
        // lane 0 -> rows0+0..7, lane 16 -> rows0+8..15
        const int base = rows0 + ((lane & 16) ? 8 : 0);
        #pragma unroll
        for (int j = 0; j < 8; ++j)
            logits[base + j] = c[j] + bias;
    }
}

// ---------------------------------------------------------------------------
// Kernel 2: one block (256 threads = 8 wave32) per span.
//  Wave 0: masked logits -> softmax over 32 positions (shuffle reductions).
//  All threads: float4-per-thread weighted sum over <=32 rows (zero-weight
//  positions skipped; exact vs reference since attn==0 contributes 0).
// ---------------------------------------------------------------------------
__global__ __launch_bounds__(256)
void span_pool_kernel(const float* __restrict__ seq,
                      const int* __restrict__ spans,
                      const float* __restrict__ logits,
                      float* __restrict__ out) {
    __shared__ float s_attn[MAXW];
    __shared__ int   s_idx[MAXW];

    const int g   = blockIdx.x;         // span id = b*N + n
    const int b   = g >> 9;             // N_ = 512
    const int tid = threadIdx.x;

    if (tid < 32) {
        const int start = spans[2 * g + 0];
        const int end   = spans[2 * g + 1];     // inclusive
        const int width = end - start;
        const int w     = tid;
        const int raw   = end - w;              // walk backward from end
        const bool valid = (w <= width) && (raw >= 0);
        const int  idx   = (raw > 0) ? raw : 0; // clamp (relu on indices)

        float lg = valid ? logits[b * S_ + idx] : MASK_FILL;

        // softmax over 32 lanes (wave32): max-subtract then normalize
        float mx = lg;
        #pragma unroll
        for (int off = 16; off > 0; off >>= 1)
            mx = fmaxf(mx, __shfl_xor(mx, off, 32));
        const float e = expf(lg - mx);          // masked -> expf(-1000-mx) == 0
        float sum = e;
        #pragma unroll
        for (int off = 16; off > 0; off >>= 1)
            sum += __shfl_xor(sum, off, 32);

        s_attn[w] = e / sum;
        s_idx[w]  = idx;
    }
    __syncthreads();

    const int d = tid * 4;                      // 256 threads * 4 = D_
    const float* base = seq + (size_t)b * S_ * D_ + d;

    float4 acc = make_float4(0.0f, 0.0f, 0.0f, 0.0f);
    #pragma unroll 4
    for (int w = 0; w < MAXW; ++w) {
        const float a = s_attn[w];
        if (a != 0.0f) {                        // uniform across block
            const float4 v = *(const float4*)(base + (size_t)s_idx[w] * D_);
            acc.x = fmaf(a, v.x, acc.x);
            acc.y = fmaf(a, v.y, acc.y);
            acc.z = fmaf(a, v.z, acc.z);
            acc.w = fmaf(a, v.w, acc.w);
        }
    }
    *(float4*)(out + (size_t)g * D_ + d) = acc;
}

// ---------------------------------------------------------------------------
// Host launcher
// Inputs (setup_inputs order): sequence_tensor f32 (B,S,D), span_indices i32
// (B,N,2), att_w f32 (D,1), att_b f32 (1). Output f32 (B,N,D).
// Workspace: logits, B*S floats = 64 KB.
// ---------------------------------------------------------------------------
extern "C" void kernel_launch(void* const* d_in, const int* in_sizes, int n_in,
                              void* d_out, int out_size, void* d_ws, size_t ws_size,
                              hipStream_t stream) {
    const float* seq   = (const float*)d_in[0];
    const int*   spans = (const int*)d_in[1];
    const float* att_w = (const float*)d_in[2];
    const float* att_b = (const float*)d_in[3];
    float*       out   = (float*)d_out;
    float*       logits = (float*)d_ws;   // B_*S_ floats = 64 KB scratch

    // Kernel 1: (B*S)/16 = 1024 waves -> 128 blocks of 8 waves
    span_logits_wmma_kernel<<<dim3((B_ * S_) / (8 * 16)), dim3(256), 0, stream>>>(
        seq, att_w, att_b, logits);

    // Kernel 2: one block per span
    span_pool_kernel<<<dim3(B_ * N_), dim3(256), 0, stream>>>(
        seq, spans, logits, out);
}